// PosLoss_34385508172504
// MI455X (gfx1250) — compile-verified
//
#include <hip/hip_runtime.h>
#include <math.h>

#define POS_NUM   60
#define ROW_F     120                      // floats per target row (2*POS_NUM)
#define BLOCK     128                      // 4 waves of 32; one row per thread
#define TWO_PI_F  6.2831853071795864769f
#define CHUNKS_PER_THREAD 30               // (BLOCK*ROW_F*4 bytes) / (16B * BLOCK)

// ---------------------------------------------------------------------------
// Kernel 1: per-row linkage loss, target tile staged via CDNA5 async LDS DMA.
// Writes deterministic per-block partial sums partial[block][5].
// ---------------------------------------------------------------------------
__global__ __launch_bounds__(BLOCK) void posloss_main(
    const float* __restrict__ pred,      // [B,5]
    const float* __restrict__ target,    // [B,120]
    float* __restrict__ partial,         // [gridDim.x,5]
    int B)
{
    __shared__ float sm[BLOCK * ROW_F];  // 60 KB staged target tile
    __shared__ float red5[5][BLOCK];     // fused category reduction scratch

    const int tid    = threadIdx.x;
    const int block0 = blockIdx.x * BLOCK;
    int rows = B - block0;
    if (rows > BLOCK) rows = BLOCK;

    // ---- Stage target tile: global -> LDS via async DMA (ASYNCcnt path) ----
    {
        const float* gbase = target + (size_t)block0 * ROW_F;
        // low 32 bits of a generic LDS pointer == byte offset within LDS
        const unsigned lds0 = (unsigned)(uintptr_t)(&sm[0]);
        if (rows == BLOCK) {
            // Full tile (the common case: B % BLOCK == 0): issue exactly 30
            // back-to-back non-temporal 16B async copies per thread, no masking.
            #pragma unroll
            for (int k = 0; k < CHUNKS_PER_THREAD; ++k) {
                unsigned boff  = ((unsigned)tid + (unsigned)k * BLOCK) * 16u;
                unsigned laddr = lds0 + boff;
                asm volatile(
                    "global_load_async_to_lds_b128 %0, %1, %2 offset:0 th:TH_LOAD_NT"
                    :: "v"(laddr), "v"(boff), "s"(gbase)
                    : "memory");
            }
        } else {
            int chunks = rows * ((ROW_F * 4) / 16);
            for (int c = tid; c < chunks; c += BLOCK) {
                unsigned boff  = (unsigned)c * 16u;
                unsigned laddr = lds0 + boff;
                asm volatile(
                    "global_load_async_to_lds_b128 %0, %1, %2 offset:0 th:TH_LOAD_NT"
                    :: "v"(laddr), "v"(boff), "s"(gbase)
                    : "memory");
            }
        }
    }

    // ---- Overlap: per-row setup math while the DMA is in flight ----
    const int  row    = block0 + tid;
    const bool active = (row < B);

    float loss = 0.0f;
    int   cat  = 0;                       // 0=invalid 1=up 2=low 3=bo 4=no
    float th2_min = 0.0f, th2_max = TWO_PI_F;
    float r1 = 2.0f, r3 = 2.0f, r4 = 2.0f, r5 = 1.0f, al = 0.0f;
    bool  validv = false;

    if (active) {
        const float* p = pred + (size_t)row * 5;
        float a0 = p[0], a1 = p[1], a2 = p[2], a3 = p[3], a4 = p[4];
        // pred9 = [r1=a0, r2=1, r3=a1, r4=a2, th1=0, r5=a3, al=a4, x0=0, y0=0]
        float L0 = a0, L1 = 1.0f, L2 = a1, L3 = a2;
        float mx  = fmaxf(fmaxf(L0, L1), fmaxf(L2, L3));
        float sum = L0 + L1 + L2 + L3;
        bool valid = (2.0f * mx < sum);
        bool c1 = (L0 + L1 - (L2 + L3)) > 0.0f;
        bool c2 = (fabsf(L0 - L1) - fabsf(L2 - L3)) >= 0.0f;
        bool up = valid &&  c1 &&  c2;
        bool lo = valid && !c1 && !c2;
        bool bo = valid &&  c1 && !c2;

        float rden = __builtin_amdgcn_rcpf(2.0f * L0 * L1);
        float s01  = L0 * L0 + L1 * L1;
        float argS = (s01 - (L2 + L3) * (L2 + L3)) * rden;
        float argD = (s01 - (L2 - L3) * (L2 - L3)) * rden;
        float acS  = acosf((up || bo) ? argS : 0.0f);
        float acD  = acosf((lo || bo) ? argD : 0.0f);

        if (valid) {
            th2_min = up ? -acS : ((lo || bo) ? acD : 0.0f);
            th2_max = (up || bo) ? acS : (lo ? (TWO_PI_F - acD) : TWO_PI_F);
            r1 = a0; r3 = a1; r4 = a2; r5 = a3; al = a4;
            cat = up ? 1 : (lo ? 2 : (bo ? 3 : 4));
        }
        validv = valid;
    }

    // Row-constant coefficients (r2=1, th1=0):
    //   k1 = K - 2*r1*cos(th2), k2 = 2*r1*r3 - 2*r3*cos(th2), b = 4*r3*sin(th2)
    const float K    = r1 * r1 + 1.0f + r3 * r3 - r4 * r4;
    const float tr1  = 2.0f * r1;
    const float tr3  = 2.0f * r3;
    const float c13  = 2.0f * r1 * r3;
    const float step = (th2_max - th2_min) * (1.0f / (POS_NUM - 1));

    // ---- DMA completion: per-wave async counter, then workgroup barrier ----
    asm volatile("s_wait_asynccnt 0x0" ::: "memory");
    __syncthreads();

    if (active) {
        const float* trow = &sm[tid * ROW_F];
        // unroll 2: two independent trans chains per iteration + the pairwise
        // ds_load_b64's of adjacent i merge into one 16B-aligned ds_load_b128
        #pragma unroll 2
        for (int i = 0; i < POS_NUM; ++i) {
            float th2 = fmaf(step, (float)i, th2_min);
            float cs  = __cosf(th2);
            float sn  = __sinf(th2);
            float k1  = fmaf(-tr1, cs, K);
            float k2  = fmaf(-tr3, cs, c13);
            float a   = k1 + k2;
            float b   = 2.0f * tr3 * sn;
            float cc  = k1 - k2;
            float disc = fmaf(b, b, -4.0f * a * cc);
            bool  bad  = disc < 0.0f;
            float D    = __builtin_amdgcn_sqrtf(bad ? 0.0f : disc);
            float t3   = 2.0f * atanf((D - b) * (0.5f * __builtin_amdgcn_rcpf(a)));
            float px   = fmaf(r5, __cosf(al + t3), cs);   // r2*cos(th2)+r5*cos(al+th3)
            float py   = fmaf(r5, __sinf(al + t3), sn);
            if (bad || !validv) { px = 0.0f; py = 0.0f; } // invalid rows -> sum(t^2)
            float dx = px - trow[2 * i];
            float dy = py - trow[2 * i + 1];
            loss = fmaf(dx, dx, loss);
            loss = fmaf(dy, dy, loss);
        }
    }

    // ---- Deterministic fused fixed-order block reduction (8 barriers) ----
    #pragma unroll
    for (int c5 = 0; c5 < 5; ++c5)
        red5[c5][tid] = (cat == c5) ? loss : 0.0f;
    __syncthreads();
    for (int off = BLOCK / 2; off > 0; off >>= 1) {
        if (tid < off) {
            #pragma unroll
            for (int c5 = 0; c5 < 5; ++c5)
                red5[c5][tid] += red5[c5][tid + off];
        }
        __syncthreads();
    }
    if (tid < 5) partial[(size_t)blockIdx.x * 5 + tid] = red5[tid][0];
}

// ---------------------------------------------------------------------------
// Kernel 2: deterministic final reduction of per-block partials -> 5 scalars.
// ---------------------------------------------------------------------------
__global__ __launch_bounds__(256) void posloss_finalize(
    const float* __restrict__ partial, int nblocks,
    float* __restrict__ out, float invB)
{
    __shared__ float red[256];
    const int tid = threadIdx.x;
    for (int c5 = 0; c5 < 5; ++c5) {
        float s = 0.0f;
        for (int b = tid; b < nblocks; b += 256)
            s += partial[(size_t)b * 5 + c5];
        red[tid] = s;
        __syncthreads();
        for (int off = 128; off > 0; off >>= 1) {
            if (tid < off) red[tid] += red[tid + off];
            __syncthreads();
        }
        if (tid == 0) out[c5] = red[0] * invB;
        __syncthreads();
    }
}

// ---------------------------------------------------------------------------
extern "C" void kernel_launch(void* const* d_in, const int* in_sizes, int n_in,
                              void* d_out, int out_size, void* d_ws, size_t ws_size,
                              hipStream_t stream)
{
    const float* pred   = (const float*)d_in[0];   // [B,5]
    const float* target = (const float*)d_in[1];   // [B,120]
    const int B = in_sizes[0] / 5;
    const int nblocks = (B + BLOCK - 1) / BLOCK;

    float* partial = (float*)d_ws;                 // nblocks*5 floats of scratch

    posloss_main<<<nblocks, BLOCK, 0, stream>>>(pred, target, partial, B);
    posloss_finalize<<<1, 256, 0, stream>>>(partial, nblocks, (float*)d_out,
                                            1.0f / (float)B);
}